// KalmanFilter_67027259621814
// MI455X (gfx1250) — compile-verified
//
#include <hip/hip_runtime.h>
#include <hip/hip_bf16.h>

typedef __attribute__((ext_vector_type(2))) float v2f;
typedef __attribute__((ext_vector_type(8))) float v8f;

constexpr int KB  = 128;   // BATCH
constexpr int KT  = 128;   // SEQ
constexpr int KD  = 64;    // D
constexpr int MSZ = KD * KD;
constexpr int NTH = 256;   // 8 wave32 waves
constexpr int NS_ITERS = 12;

// ---- 16x16 f32 C/D tile load/store, ISA 7.12.2 layout:
//      VGPR v: lanes 0-15 -> M=v, N=lane ; lanes 16-31 -> M=v+8, N=lane-16
__device__ __forceinline__ v8f load_tile(const float* __restrict__ M, int tm, int tn) {
    const int l = threadIdx.x & 31;
    const int r = l & 15;
    const int half = l >> 4;
    const int n = tn * 16 + r;
    v8f c;
#pragma unroll
    for (int v = 0; v < 8; ++v) c[v] = M[(tm * 16 + half * 8 + v) * KD + n];
    return c;
}

__device__ __forceinline__ void store_tile(float* __restrict__ M, int tm, int tn, v8f c) {
    const int l = threadIdx.x & 31;
    const int r = l & 15;
    const int half = l >> 4;
    const int n = tn * 16 + r;
#pragma unroll
    for (int v = 0; v < 8; ++v) M[(tm * 16 + half * 8 + v) * KD + n] = c[v];
}

// ---- one 16x16 output tile of a 64x64 matmul, K swept in steps of 4 via
//      V_WMMA_F32_16X16X4_F32. A-frag: lanes 0-15 hold K=k0..k0+1 (.x,.y),
//      lanes 16-31 hold K=k0+2..k0+3. B-frag mirrored (row-striped).
template <int At, int Bt>
__device__ __forceinline__ v8f wmma_mm64_tile(const float* __restrict__ A,
                                              const float* __restrict__ B,
                                              int tm, int tn) {
    v8f acc = {};
    const int l = threadIdx.x & 31;
    const int r = l & 15;
    const int h = (l >> 4) << 1;        // 0 or 2
    const int m = tm * 16 + r;
    const int n = tn * 16 + r;
#pragma unroll
    for (int k0 = 0; k0 < KD; k0 += 4) {
        const int ka = k0 + h;
        v2f a, b;
        if (At) { a.x = A[ka * KD + m];       a.y = A[(ka + 1) * KD + m]; }
        else    { a.x = A[m * KD + ka];       a.y = A[m * KD + ka + 1];   }
        if (Bt) { b.x = B[n * KD + ka];       b.y = B[n * KD + ka + 1];   }
        else    { b.x = B[ka * KD + n];       b.y = B[(ka + 1) * KD + n]; }
        acc = __builtin_amdgcn_wmma_f32_16x16x4_f32(false, a, false, b,
                                                    (short)0, acc, false, false);
    }
    return acc;
}

// ---- full 64x64 matmul: Dst = op(A) * op(B) [+ Cadd].  8 waves x 2 tiles.
//      Dst must not alias A/B/Cadd.  Trailing barrier publishes Dst.
template <int At, int Bt, bool ADD>
__device__ __forceinline__ void mm64(float* __restrict__ Dst,
                                     const float* __restrict__ A,
                                     const float* __restrict__ B,
                                     const float* __restrict__ Cadd) {
    const int w  = threadIdx.x >> 5;
    const int t0 = 2 * w, t1 = t0 + 1;
    v8f a0 = wmma_mm64_tile<At, Bt>(A, B, t0 >> 2, t0 & 3);
    v8f a1 = wmma_mm64_tile<At, Bt>(A, B, t1 >> 2, t1 & 3);
    if (ADD) {
        a0 = a0 + load_tile(Cadd, t0 >> 2, t0 & 3);
        a1 = a1 + load_tile(Cadd, t1 >> 2, t1 & 3);
    }
    store_tile(Dst, t0 >> 2, t0 & 3, a0);
    store_tile(Dst, t1 >> 2, t1 & 3, a1);
    __syncthreads();
}

__global__ void __launch_bounds__(NTH)
KalmanFilter_67027259621814_kernel(const float* __restrict__ x0,
                                   const float* __restrict__ u,
                                   const float* __restrict__ yy,
                                   const float* __restrict__ A,
                                   const float* __restrict__ Bm,
                                   const float* __restrict__ C,
                                   const float* __restrict__ LNs,
                                   const float* __restrict__ LNo,
                                   float* __restrict__ pred_out,
                                   float* __restrict__ cov_out) {
    extern __shared__ float smem[];
    float* As  = smem;              // A (resident)
    float* Cs  = As  + MSZ;         // C (resident)
    float* P   = Cs  + MSZ;         // state covariance (resident across steps)
    float* Nss = P   + MSZ;         // Ns = LNs LNs^T (resident)
    float* Noo = Nss + MSZ;         // No = LNo LNo^T (resident)
    float* Pp  = Noo + MSZ;         // P_pred
    float* TMP = Pp  + MSZ;         // T1, then CP = C*P_pred
    float* Sm  = TMP + MSZ;         // S, later G
    float* X   = Sm  + MSZ;         // S_inv (Newton-Schulz), later G*CP
    float* Ym  = X   + MSZ;         // NS temp, later P_pred*C^T
    float* xv  = Ym  + MSZ;         // state mean [64]
    float* xp  = xv  + KD;          // x_pred   [64]
    float* iv  = xp  + KD;          // innov    [64]
    float* red = iv  + KD;          // row sums [64] + alpha at red[64]

    const int b   = blockIdx.x;
    const int tid = threadIdx.x;

    // ---- init: params into LDS, P = I, x
    for (int i = tid; i < MSZ; i += NTH) {
        As[i] = A[i];
        Cs[i] = C[i];
        P[i]  = ((i >> 6) == (i & 63)) ? 1.0f : 0.0f;
    }
    if (tid < KD) xv[tid] = x0[b * KD + tid];
    __syncthreads();

    // ---- Ns = LNs LNs^T,  No = LNo LNo^T (once per block)
    for (int i = tid; i < MSZ; i += NTH) TMP[i] = LNs[i];
    __syncthreads();
    mm64<0, 1, false>(Nss, TMP, TMP, nullptr);
    for (int i = tid; i < MSZ; i += NTH) TMP[i] = LNo[i];
    __syncthreads();
    mm64<0, 1, false>(Noo, TMP, TMP, nullptr);

    for (int t = 0; t < KT; ++t) {
        const float* ut = u  + ((size_t)b * KT + t) * KD;
        const float* yt = yy + ((size_t)b * KT + t) * KD;

        // prefetch next step's u/y rows while we crunch this step
        if (t + 1 < KT && tid < KD) {
            __builtin_prefetch(ut + KD + tid, 0, 0);
            __builtin_prefetch(yt + KD + tid, 0, 0);
        }

        // ---- mean: x_pred = A x + Bm u_t ;  innov = y_t - C x_pred
        if (tid < KD) {
            float s = 0.f;
            for (int j = 0; j < KD; ++j) s += As[tid * KD + j] * xv[j] + Bm[tid * KD + j] * ut[j];
            xp[tid] = s;
        }
        __syncthreads();
        if (tid < KD) {
            float s = 0.f;
            for (int j = 0; j < KD; ++j) s += Cs[tid * KD + j] * xp[j];
            iv[tid] = yt[tid] - s;
        }
        __syncthreads();

        // ---- covariance pipeline (all WMMA f32)
        mm64<0, 0, false>(TMP, As, P,  nullptr);  // T1 = A*P
        mm64<0, 1, true >(Pp,  TMP, As, Nss);     // Pp = T1*A^T + Ns
        mm64<0, 0, false>(TMP, Cs, Pp, nullptr);  // CP = C*Pp
        mm64<0, 1, true >(Sm,  TMP, Cs, Noo);     // S  = CP*C^T + No

        // ---- S_inv via Newton-Schulz: X0 = S / (||S||_1 ||S||_inf)
        if (tid < KD) {
            float s = 0.f;
            for (int j = 0; j < KD; ++j) s += fabsf(Sm[tid * KD + j]);
            red[tid] = s;
        }
        __syncthreads();
        if (tid == 0) {
            float mx = 0.f;
            for (int j = 0; j < KD; ++j) mx = fmaxf(mx, red[j]);
            red[KD] = 1.0f / (mx * mx);           // S symmetric: ||S||_1 == ||S||_inf
        }
        __syncthreads();
        {
            const float alpha = red[KD];
            for (int i = tid; i < MSZ; i += NTH) X[i] = alpha * Sm[i];
        }
        __syncthreads();

        for (int it = 0; it < NS_ITERS; ++it) {
            mm64<0, 0, false>(Ym, Sm, X, nullptr);          // Y = S*X
            // X = 2X - X*Y   (compute in regs, barrier, then in-place update)
            const int w  = tid >> 5;
            const int t0 = 2 * w, t1 = t0 + 1;
            v8f a0 = wmma_mm64_tile<0, 0>(X, Ym, t0 >> 2, t0 & 3);
            v8f a1 = wmma_mm64_tile<0, 0>(X, Ym, t1 >> 2, t1 & 3);
            __syncthreads();                                 // all reads of X done
            v8f o0 = load_tile(X, t0 >> 2, t0 & 3);
            v8f o1 = load_tile(X, t1 >> 2, t1 & 3);
            store_tile(X, t0 >> 2, t0 & 3, o0 * 2.0f - a0);
            store_tile(X, t1 >> 2, t1 & 3, o1 * 2.0f - a1);
            __syncthreads();
        }

        mm64<0, 1, false>(Ym, Pp, Cs, nullptr);   // T2 = Pp*C^T   (Y slot free)
        mm64<0, 0, false>(Sm, Ym, X,  nullptr);   // G  = T2*S_inv (S slot free)

        // ---- mean update + prediction output
        if (tid < KD) {
            float s = xp[tid];
            for (int j = 0; j < KD; ++j) s += Sm[tid * KD + j] * iv[j];
            xv[tid] = s;
            __builtin_nontemporal_store(s, pred_out + ((size_t)b * KT + t) * KD + tid);
        }
        __syncthreads();

        mm64<0, 0, false>(X, Sm, TMP, nullptr);   // GCP = G*CP  (X slot free)

        // ---- P = 0.5(Pp+Pp^T) - 0.5(GCP+GCP^T); stream covariance out (NT)
        {
            float* cov = cov_out + ((size_t)b * KT + t) * MSZ;
#pragma unroll
            for (int e = 0; e < MSZ / NTH; ++e) {
                const int idx = e * NTH + tid;
                const int i = idx >> 6, j = idx & 63;
                const float vP = 0.5f * (Pp[idx] + Pp[j * KD + i])
                               - 0.5f * (X[idx]  + X[j * KD + i]);
                P[idx] = vP;
                __builtin_nontemporal_store(vP, cov + idx);
            }
        }
        __syncthreads();
    }
}

extern "C" void kernel_launch(void* const* d_in, const int* in_sizes, int n_in,
                              void* d_out, int out_size, void* d_ws, size_t ws_size,
                              hipStream_t stream) {
    const float* x0  = (const float*)d_in[0];
    const float* u   = (const float*)d_in[1];
    const float* yy  = (const float*)d_in[2];
    const float* A   = (const float*)d_in[3];
    const float* Bm  = (const float*)d_in[4];
    const float* C   = (const float*)d_in[5];
    const float* LNs = (const float*)d_in[6];
    const float* LNo = (const float*)d_in[7];

    float* pred = (float*)d_out;                          // [B,T,D]
    float* cov  = pred + (size_t)KB * KT * KD;            // [B,T,D,D]

    const size_t smem_bytes = (size_t)(10 * MSZ + 4 * KD + 16) * sizeof(float); // ~165 KB of 320 KB/WGP

    KalmanFilter_67027259621814_kernel<<<dim3(KB), dim3(NTH), smem_bytes, stream>>>(
        x0, u, yy, A, Bm, C, LNs, LNo, pred, cov);
}